// MPNN_84610855731437
// MI455X (gfx1250) — compile-verified
//
#include <hip/hip_runtime.h>
#include <math.h>

// ---------------- problem constants (match reference) ----------------
#define NN   4096      // nodes
#define NF   128       // input features
#define DD   64        // hidden dim
#define GHD  64        // edge-net hidden
#define EE   32768     // edges
#define BB   32        // graphs
#define NG   3
#define STEPS 3
#define KTOT 4160      // 64*64 bilinear K + 64 bias rows

typedef __attribute__((ext_vector_type(2))) float v2f;
typedef __attribute__((ext_vector_type(8))) float v8f;

__device__ __forceinline__ float relu_(float x) { return x > 0.f ? x : 0.f; }
__device__ __forceinline__ float sig_(float x)  { return 1.f / (1.f + expf(-x)); }

// ---------------- generic fill ----------------
__global__ void fill_f32(float* p, float v, int n) {
    int i = blockIdx.x * blockDim.x + threadIdx.x;
    if (i < n) p[i] = v;
}

// ---------------- degree count ----------------
__global__ void deg_count(const int* __restrict__ dst, float* __restrict__ deg) {
    int e = blockIdx.x * blockDim.x + threadIdx.x;
    if (e < EE) atomicAdd(&deg[dst[e]], 1.0f);
}

// ---------------- repack enn_W2/b2 into W2p (KTOT x 64) ----------------
// W2p[g*64+i][o] = W2[g][i*64+o];  W2p[4096+i][o] = b2[i*64+o]
__global__ void repack_w2(const float* __restrict__ W2, const float* __restrict__ b2,
                          float* __restrict__ W2p)
{
    int idx = blockIdx.x * blockDim.x + threadIdx.x;   // KTOT*64
    if (idx >= KTOT * DD) return;
    int k = idx >> 6, o = idx & 63;
    if (k < 4096) {
        int g = k >> 6, i = k & 63;
        W2p[idx] = W2[g * 4096 + i * DD + o];
    } else {
        int i = k - 4096;
        W2p[idx] = b2[i * DD + o];
    }
}

// =====================================================================
// Fused edge-message GEMM (f32 WMMA), 128 edges / block.
// msg(128x64) = A(128xKTOT) @ W2p(KTOTx64),
//   A[e, g*64+i] = h1[e,g]*x[src_e,i]   (bias rows: A[e,4096+i]=x[src_e,i])
// built on the fly in registers from LDS-resident x-rows and h1.
// Each of 8 waves computes a 32x32 output tile (2 Mtiles x 2 Ntiles):
// B fragments reused across M, A fragments across N -> 4 WMMA per
// {2 A b64 + 2 B b64 + 2 pk_mul}. Epilogue scatters with atomicAdd(agg[dst]).
// =====================================================================
__global__ __launch_bounds__(256) void edge_gemm(
    const float* __restrict__ hnode, const float* __restrict__ ea,
    const int*   __restrict__ eidx,  const float* __restrict__ eW1,
    const float* __restrict__ eb1,   const float* __restrict__ W2p,
    float* __restrict__ agg)
{
    __shared__ __align__(16) float lx[128][66];  // gathered out[src] rows
    __shared__ __align__(16) float lh[128][66];  // edge hidden h1
    __shared__ __align__(16) float lB[64][66];   // B panel transposed: lB[col][k_local]
    __shared__ int lsrc[128];
    __shared__ int ldst[128];

    const int tid = threadIdx.x;
    const int e0  = blockIdx.x * 128;

    if (tid < 128) {
        lsrc[tid] = eidx[e0 + tid];
        ldst[tid] = eidx[EE + e0 + tid];
    }
    __syncthreads();

    // gather x rows (64 floats each) and compute h1 = relu(ea*W1+b1)
    {
        int e = tid >> 1, c0 = (tid & 1) * 32;
        const float* srcp = hnode + (size_t)lsrc[e] * DD + c0;
        #pragma unroll
        for (int q = 0; q < 32; q += 4) {
            float4 v = *(const float4*)(srcp + q);
            lx[e][c0 + q]     = v.x;
            lx[e][c0 + q + 1] = v.y;
            lx[e][c0 + q + 2] = v.z;
            lx[e][c0 + q + 3] = v.w;
        }
        float eav = ea[e0 + e];
        #pragma unroll
        for (int q = 0; q < 32; ++q) {
            int g = c0 + q;
            lh[e][g] = relu_(eav * eW1[g] + eb1[g]);
        }
    }

    const int lane  = tid & 31;
    const int w     = tid >> 5;          // 8 waves
    const int mt0   = (w & 3) * 2;       // two M tiles: mt0, mt0+1
    const int ncol0 = (w >> 2) * 32;     // two N tiles: ncol0, ncol0+16
    const int half  = lane >> 4;
    const int lrow  = lane & 15;
    const int row0  = mt0 * 16 + lrow;   // A rows for this lane
    const int row1  = row0 + 16;
    const int klo   = half * 2;          // lanes0-15: K{0,1}, lanes16-31: K{2,3}

    v8f c00 = {0.f,0.f,0.f,0.f,0.f,0.f,0.f,0.f};
    v8f c01 = {0.f,0.f,0.f,0.f,0.f,0.f,0.f,0.f};
    v8f c10 = {0.f,0.f,0.f,0.f,0.f,0.f,0.f,0.f};
    v8f c11 = {0.f,0.f,0.f,0.f,0.f,0.f,0.f,0.f};

    for (int kc = 0; kc < KTOT / 64; ++kc) {   // 65 K-panels of 64
        __syncthreads();
        // load B panel (64x64) transposed into LDS
        {
            const float* wsrc = W2p + (size_t)kc * 64 * DD;
            int kl = tid >> 2, cb = (tid & 3) * 16;
            #pragma unroll
            for (int j = 0; j < 16; j += 4) {
                float4 v = *(const float4*)(wsrc + kl * DD + cb + j);
                lB[cb + j][kl]     = v.x;
                lB[cb + j + 1][kl] = v.y;
                lB[cb + j + 2][kl] = v.z;
                lB[cb + j + 3][kl] = v.w;
            }
        }
        __syncthreads();

        const bool isbias = (kc >= 64);
        const float hv0 = isbias ? 1.0f : lh[row0][kc];   // g == kc within panel
        const float hv1 = isbias ? 1.0f : lh[row1][kc];
        #pragma unroll
        for (int kk = 0; kk < 64; kk += 4) {
            v2f xv0 = *(const v2f*)&lx[row0][kk + klo];
            v2f xv1 = *(const v2f*)&lx[row1][kk + klo];
            v2f a0; a0.x = hv0 * xv0.x; a0.y = hv0 * xv0.y;
            v2f a1; a1.x = hv1 * xv1.x; a1.y = hv1 * xv1.y;
            v2f b0 = *(const v2f*)&lB[ncol0 + lrow][kk + klo];
            v2f b1 = *(const v2f*)&lB[ncol0 + 16 + lrow][kk + klo];
            c00 = __builtin_amdgcn_wmma_f32_16x16x4_f32(false, a0, false, b0, (short)0, c00, false, false);
            c01 = __builtin_amdgcn_wmma_f32_16x16x4_f32(false, a0, false, b1, (short)0, c01, false, false);
            c10 = __builtin_amdgcn_wmma_f32_16x16x4_f32(false, a1, false, b0, (short)0, c10, false, false);
            c11 = __builtin_amdgcn_wmma_f32_16x16x4_f32(false, a1, false, b1, (short)0, c11, false, false);
        }
    }

    // epilogue: C layout: VGPR j -> M = 8*half + j, N = lrow
    #pragma unroll
    for (int j = 0; j < 8; ++j) {
        int mA = mt0 * 16 + half * 8 + j;
        int mB = mA + 16;
        int dA = ldst[mA];
        int dB = ldst[mB];
        atomicAdd(&agg[(size_t)dA * DD + ncol0 + lrow],      c00[j]);
        atomicAdd(&agg[(size_t)dA * DD + ncol0 + 16 + lrow], c01[j]);
        atomicAdd(&agg[(size_t)dB * DD + ncol0 + lrow],      c10[j]);
        atomicAdd(&agg[(size_t)dB * DD + ncol0 + 16 + lrow], c11[j]);
    }
}

// =====================================================================
// Generic WMMA node GEMM: out(M x Ncols) = epi( A(M x K) @ W(K x Ncols) + bias )
// 64 rows per block, K in {64,128}, Ncols in {64,192}.
// mode 0: linear; mode 1: relu; mode 2: relu( . + agg[r,c]/max(deg[r],1) )
// =====================================================================
__global__ __launch_bounds__(256) void node_gemm(
    const float* __restrict__ A, const float* __restrict__ W,
    const float* __restrict__ bias,
    const float* __restrict__ agg, const float* __restrict__ deg,
    float* __restrict__ out, int K, int Ncols, int mode)
{
    __shared__ __align__(16) float lA[64 * 130];   // rows, stride SW=K+2
    __shared__ __align__(16) float wT[12672];      // wT[c*SW + k], worst 192*66

    const int tid  = threadIdx.x;
    const int row0 = blockIdx.x * 64;
    const int SW   = K + 2;
    const int K4   = K >> 2;

    // stage A rows
    for (int i = tid; i < 64 * K4; i += 256) {
        int r = i / K4, c4 = (i - r * K4) * 4;
        float4 v = *(const float4*)(A + (size_t)(row0 + r) * K + c4);
        float* dp = &lA[r * SW + c4];
        dp[0] = v.x; dp[1] = v.y; dp[2] = v.z; dp[3] = v.w;
    }
    // stage W transposed
    for (int i = tid; i < K * Ncols; i += 256) {
        int k = i / Ncols, c = i - k * Ncols;
        wT[c * SW + k] = W[i];
    }
    __syncthreads();

    const int lane  = tid & 31;
    const int w     = tid >> 5;
    const int mtile = w & 3;
    const int half  = lane >> 4;
    const int lrow  = lane & 15;
    const int row   = mtile * 16 + lrow;
    const int klo   = half * 2;
    const int ntiles = Ncols >> 4;

    const float* ap = &lA[row * SW];
    for (int nt = (w >> 2); nt < ntiles; nt += 2) {
        const float* bp = &wT[(nt * 16 + lrow) * SW];
        v8f acc = {0.f,0.f,0.f,0.f,0.f,0.f,0.f,0.f};
        for (int kk = 0; kk < K; kk += 4) {
            v2f a = *(const v2f*)(ap + kk + klo);
            v2f b = *(const v2f*)(bp + kk + klo);
            acc = __builtin_amdgcn_wmma_f32_16x16x4_f32(false, a, false, b, (short)0, acc, false, false);
        }
        int c = nt * 16 + lrow;
        float bv = bias[c];
        #pragma unroll
        for (int j = 0; j < 8; ++j) {
            int r = row0 + mtile * 16 + half * 8 + j;
            float val = acc[j] + bv;
            if (mode == 2) val += agg[(size_t)r * DD + c] / fmaxf(deg[r], 1.0f);
            if (mode >= 1) val = relu_(val);
            out[(size_t)r * Ncols + c] = val;
        }
    }
}

// ---------------- GRU gates, in-place h update ----------------
__global__ void gru_gate(const float* __restrict__ gi, const float* __restrict__ gh,
                         float* __restrict__ h)
{
    int idx = blockIdx.x * blockDim.x + threadIdx.x;
    if (idx >= NN * DD) return;
    int n = idx >> 6, d = idx & 63;
    size_t base = (size_t)n * 3 * DD;
    float r  = sig_(gi[base + d]        + gh[base + d]);
    float z  = sig_(gi[base + DD + d]   + gh[base + DD + d]);
    float nn = tanhf(gi[base + 2*DD + d] + r * gh[base + 2*DD + d]);
    float ho = h[idx];
    h[idx] = (1.f - z) * nn + z * ho;
}

// ---------------- set2set: LSTM gates ----------------
__global__ __launch_bounds__(256) void s2s_gates(
    const float* __restrict__ qstar, const float* __restrict__ hh,
    const float* __restrict__ Wih, const float* __restrict__ Whh,
    const float* __restrict__ bih, const float* __restrict__ bhh,
    float* __restrict__ gates)
{
    int b = blockIdx.x, j = threadIdx.x;   // 32 x 256
    float acc = bih[j] + bhh[j];
    const float* q = qstar + (size_t)b * 2 * DD;
    const float* h = hh    + (size_t)b * DD;
    for (int k = 0; k < 2 * DD; ++k) acc += q[k] * Wih[k * 4 * DD + j];
    for (int k = 0; k < DD; ++k)     acc += h[k] * Whh[k * 4 * DD + j];
    gates[(size_t)b * 4 * DD + j] = acc;
}

// ---------------- set2set: LSTM cell update (in place) ----------------
__global__ void s2s_lstm(const float* __restrict__ gates,
                         float* __restrict__ hh, float* __restrict__ cc)
{
    int idx = blockIdx.x * blockDim.x + threadIdx.x;
    if (idx >= BB * DD) return;
    int b = idx >> 6, d = idx & 63;
    size_t g = (size_t)b * 4 * DD;
    float gi = gates[g + d];
    float gf = gates[g + DD + d];
    float gg = gates[g + 2*DD + d];
    float go = gates[g + 3*DD + d];
    float c = sig_(gf) * cc[idx] + sig_(gi) * tanhf(gg);
    cc[idx] = c;
    hh[idx] = sig_(go) * tanhf(c);
}

__device__ __forceinline__ void atomicMaxF(float* addr, float v) {
    if (v >= 0.f) atomicMax((int*)addr, __float_as_int(v));
    else          atomicMin((unsigned int*)addr, __float_as_uint(v));
}

// ---------------- e = dot(out, q[batch]) ; segment max ----------------
__global__ void s2s_e(const float* __restrict__ h, const float* __restrict__ hh,
                      const int* __restrict__ bmap, float* __restrict__ e,
                      float* __restrict__ m)
{
    int i = blockIdx.x * blockDim.x + threadIdx.x;
    if (i >= NN) return;
    int b = bmap[i];
    const float* hr = h + (size_t)i * DD;
    const float* q  = hh + (size_t)b * DD;
    float s = 0.f;
    for (int k = 0; k < DD; ++k) s += hr[k] * q[k];
    e[i] = s;
    atomicMaxF(&m[b], s);
}

// ---------------- a = exp(e-m[b]) ; segment sum den ----------------
__global__ void s2s_a(const float* __restrict__ e, const float* __restrict__ m,
                      const int* __restrict__ bmap, float* __restrict__ a,
                      float* __restrict__ den)
{
    int i = blockIdx.x * blockDim.x + threadIdx.x;
    if (i >= NN) return;
    int b = bmap[i];
    float av = expf(e[i] - m[b]);
    a[i] = av;
    atomicAdd(&den[b], av);
}

// ---------------- r_read += (a/den) * out ----------------
__global__ void s2s_r(const float* __restrict__ h, const float* __restrict__ a,
                      const float* __restrict__ den, const int* __restrict__ bmap,
                      float* __restrict__ rread)
{
    int idx = blockIdx.x * blockDim.x + threadIdx.x;
    if (idx >= NN * DD) return;
    int i = idx >> 6, d = idx & 63;
    int b = bmap[i];
    atomicAdd(&rread[(size_t)b * DD + d], (a[i] / den[b]) * h[idx]);
}

// ---------------- q_star = concat(hh, r_read) ----------------
__global__ void s2s_q(const float* __restrict__ hh, const float* __restrict__ rread,
                      float* __restrict__ qstar)
{
    int idx = blockIdx.x * blockDim.x + threadIdx.x;
    if (idx >= BB * 2 * DD) return;
    int b = idx >> 7, j = idx & 127;
    qstar[idx] = (j < DD) ? hh[(size_t)b * DD + j] : rread[(size_t)b * DD + j - DD];
}

// ---------------- post MLP: (B,128)->(B,1) ----------------
__global__ __launch_bounds__(64) void post_mlp(
    const float* __restrict__ qstar,
    const float* __restrict__ W0, const float* __restrict__ b0,
    const float* __restrict__ W1, const float* __restrict__ b1,
    const float* __restrict__ W2, const float* __restrict__ b2,
    const float* __restrict__ W3, const float* __restrict__ b3,
    float* __restrict__ y)
{
    __shared__ float si[2 * DD];
    __shared__ float s0[DD];
    __shared__ float s1[DD];
    __shared__ float s2[DD];
    int b = blockIdx.x, t = threadIdx.x;
    si[t]      = qstar[(size_t)b * 2 * DD + t];
    si[t + 64] = qstar[(size_t)b * 2 * DD + t + 64];
    __syncthreads();
    float acc = b0[t];
    for (int k = 0; k < 2 * DD; ++k) acc += si[k] * W0[k * DD + t];
    s0[t] = relu_(acc);
    __syncthreads();
    acc = b1[t];
    for (int k = 0; k < DD; ++k) acc += s0[k] * W1[k * DD + t];
    s1[t] = relu_(acc);
    __syncthreads();
    acc = b2[t];
    for (int k = 0; k < DD; ++k) acc += s1[k] * W2[k * DD + t];
    s2[t] = relu_(acc);
    __syncthreads();
    if (t == 0) {
        float o = b3[0];
        for (int k = 0; k < DD; ++k) o += s2[k] * W3[k];
        y[b] = o;
    }
}

// ==================== launch ====================
extern "C" void kernel_launch(void* const* d_in, const int* in_sizes, int n_in,
                              void* d_out, int out_size, void* d_ws, size_t ws_size,
                              hipStream_t stream)
{
    const float* x        = (const float*)d_in[0];
    const float* eattr    = (const float*)d_in[1];
    const int*   eidx     = (const int*)  d_in[2];
    const int*   bmap     = (const int*)  d_in[3];
    const float* pre_W0   = (const float*)d_in[4];
    const float* pre_b0   = (const float*)d_in[5];
    const float* pre_W1   = (const float*)d_in[6];
    const float* pre_b1   = (const float*)d_in[7];
    const float* pre_W2   = (const float*)d_in[8];
    const float* pre_b2   = (const float*)d_in[9];
    const float* enn_W1   = (const float*)d_in[10];
    const float* enn_b1   = (const float*)d_in[11];
    const float* enn_W2   = (const float*)d_in[12];
    const float* enn_b2   = (const float*)d_in[13];
    const float* root_W   = (const float*)d_in[14];
    const float* conv_b   = (const float*)d_in[15];
    const float* gru_Wih  = (const float*)d_in[16];
    const float* gru_Whh  = (const float*)d_in[17];
    const float* gru_bih  = (const float*)d_in[18];
    const float* gru_bhh  = (const float*)d_in[19];
    const float* s2s_Wih  = (const float*)d_in[20];
    const float* s2s_Whh  = (const float*)d_in[21];
    const float* s2s_bih  = (const float*)d_in[22];
    const float* s2s_bhh  = (const float*)d_in[23];
    const float* post_W0  = (const float*)d_in[24];
    const float* post_b0  = (const float*)d_in[25];
    const float* post_W1  = (const float*)d_in[26];
    const float* post_b1  = (const float*)d_in[27];
    const float* post_W2  = (const float*)d_in[28];
    const float* post_b2  = (const float*)d_in[29];
    const float* post_W3  = (const float*)d_in[30];
    const float* post_b3  = (const float*)d_in[31];
    float* y = (float*)d_out;

    // workspace layout (floats)
    float* ws     = (float*)d_ws;
    float* h_node = ws;                       // N*D
    float* agg    = h_node + NN * DD;         // N*D
    float* convb  = agg    + NN * DD;         // N*D
    float* gi     = convb  + NN * DD;         // N*192 (also pre-MLP tmp)
    float* gh     = gi     + NN * 3 * DD;     // N*192 (also pre-MLP tmp)
    float* deg    = gh     + NN * 3 * DD;     // N
    float* W2p    = deg    + NN;              // KTOT*64
    float* e_buf  = W2p    + KTOT * DD;       // N
    float* a_buf  = e_buf  + NN;              // N
    float* gates  = a_buf  + NN;              // 32*256
    float* qstar  = gates  + BB * 4 * DD;     // 32*128
    float* hh     = qstar  + BB * 2 * DD;     // 32*64
    float* cc     = hh     + BB * DD;         // 32*64
    float* m_buf  = cc     + BB * DD;         // 32
    float* den    = m_buf  + BB;              // 32
    float* rread  = den    + BB;              // 32*64

    // 1. pre MLP (three WMMA GEMMs; gi/gh used as temporaries)
    node_gemm<<<NN / 64, 256, 0, stream>>>(x,  pre_W0, pre_b0, nullptr, nullptr, gi,     NF, DD, 1);
    node_gemm<<<NN / 64, 256, 0, stream>>>(gi, pre_W1, pre_b1, nullptr, nullptr, gh,     DD, DD, 1);
    node_gemm<<<NN / 64, 256, 0, stream>>>(gh, pre_W2, pre_b2, nullptr, nullptr, h_node, DD, DD, 1);

    // 2. degree
    fill_f32<<<(NN + 255) / 256, 256, 0, stream>>>(deg, 0.f, NN);
    deg_count<<<EE / 256, 256, 0, stream>>>(eidx + EE, deg);

    // 3. NG message-passing layers
    for (int l = 0; l < NG; ++l) {
        repack_w2<<<(KTOT * DD) / 256, 256, 0, stream>>>(
            enn_W2 + (size_t)l * GHD * DD * DD, enn_b2 + (size_t)l * DD * DD, W2p);
        fill_f32<<<(NN * DD) / 256, 256, 0, stream>>>(agg, 0.f, NN * DD);
        edge_gemm<<<EE / 128, 256, 0, stream>>>(
            h_node, eattr, eidx, enn_W1 + (size_t)l * GHD, enn_b1 + (size_t)l * GHD, W2p, agg);
        // conv = relu(agg/deg + h@rootW + cb)
        node_gemm<<<NN / 64, 256, 0, stream>>>(
            h_node, root_W + (size_t)l * DD * DD, conv_b + (size_t)l * DD, agg, deg, convb, DD, DD, 2);
        // GRU matmuls
        node_gemm<<<NN / 64, 256, 0, stream>>>(
            convb, gru_Wih + (size_t)l * DD * 3 * DD, gru_bih + (size_t)l * 3 * DD,
            nullptr, nullptr, gi, DD, 3 * DD, 0);
        node_gemm<<<NN / 64, 256, 0, stream>>>(
            h_node, gru_Whh + (size_t)l * DD * 3 * DD, gru_bhh + (size_t)l * 3 * DD,
            nullptr, nullptr, gh, DD, 3 * DD, 0);
        gru_gate<<<(NN * DD) / 256, 256, 0, stream>>>(gi, gh, h_node);
    }

    // 4. set2set init: qstar, hh, cc contiguous -> one zero fill
    fill_f32<<<(BB * (2*DD + DD + DD) + 255) / 256, 256, 0, stream>>>(
        qstar, 0.f, BB * (2 * DD + DD + DD));

    for (int s = 0; s < STEPS; ++s) {
        s2s_gates<<<BB, 4 * DD, 0, stream>>>(qstar, hh, s2s_Wih, s2s_Whh, s2s_bih, s2s_bhh, gates);
        s2s_lstm<<<(BB * DD + 255) / 256, 256, 0, stream>>>(gates, hh, cc);
        fill_f32<<<1, BB, 0, stream>>>(m_buf, -INFINITY, BB);
        fill_f32<<<(BB + BB * DD + 255) / 256, 256, 0, stream>>>(den, 0.f, BB + BB * DD);
        s2s_e<<<NN / 256, 256, 0, stream>>>(h_node, hh, bmap, e_buf, m_buf);
        s2s_a<<<NN / 256, 256, 0, stream>>>(e_buf, m_buf, bmap, a_buf, den);
        s2s_r<<<(NN * DD) / 256, 256, 0, stream>>>(h_node, a_buf, den, bmap, rread);
        s2s_q<<<(BB * 2 * DD + 255) / 256, 256, 0, stream>>>(hh, rread, qstar);
    }

    // 5. post MLP -> y (B,1)
    post_mlp<<<BB, 64, 0, stream>>>(qstar, post_W0, post_b0, post_W1, post_b1,
                                    post_W2, post_b2, post_W3, post_b3, y);
    (void)in_sizes; (void)n_in; (void)out_size; (void)ws_size;
}